// InertiaOfBeliefMass_77704548319435
// MI455X (gfx1250) — compile-verified
//
#include <hip/hip_runtime.h>
#include <math.h>

#define EPS_REG 1e-6f
#define MIN_EIG 1e-4f

typedef float v2f __attribute__((ext_vector_type(2)));
typedef float v8f __attribute__((ext_vector_type(8)));

// ---------- small 3x3 helpers ----------
__device__ __forceinline__ void inv3(const float* m, float* out) {
  float a=m[0],b=m[1],c=m[2],d=m[3],e=m[4],f=m[5],g=m[6],h=m[7],i=m[8];
  float A = e*i - f*h;
  float B = f*g - d*i;
  float C = d*h - e*g;
  float det = a*A + b*B + c*C;
  float r = 1.0f / det;
  out[0]=A*r;          out[1]=(c*h-b*i)*r;  out[2]=(b*f-c*e)*r;
  out[3]=B*r;          out[4]=(a*i-c*g)*r;  out[5]=(c*d-a*f)*r;
  out[6]=C*r;          out[7]=(b*g-a*h)*r;  out[8]=(a*e-b*d)*r;
}

__device__ __forceinline__ void mm3(const float* A, const float* B, float* C) { // C = A*B
  for (int r = 0; r < 3; ++r)
    for (int c = 0; c < 3; ++c)
      C[3*r+c] = A[3*r+0]*B[0*3+c] + A[3*r+1]*B[1*3+c] + A[3*r+2]*B[2*3+c];
}
__device__ __forceinline__ void mTm3(const float* A, const float* B, float* C) { // C = A^T*B
  for (int r = 0; r < 3; ++r)
    for (int c = 0; c < 3; ++c)
      C[3*r+c] = A[0*3+r]*B[0*3+c] + A[1*3+r]*B[1*3+c] + A[2*3+r]*B[2*3+c];
}
__device__ __forceinline__ void mmT3(const float* A, const float* B, float* C) { // C = A*B^T
  for (int r = 0; r < 3; ++r)
    for (int c = 0; c < 3; ++c)
      C[3*r+c] = A[3*r+0]*B[3*c+0] + A[3*r+1]*B[3*c+1] + A[3*r+2]*B[3*c+2];
}

// ---------- kernel 1: per-site precompute ----------
// Lambda_p = inv(Sp + eps I), Lambda_q = inv(Sq + eps I),
// R = expm(hat(phi)) via Rodrigues, A = R^T Lq R  (stored padded to 16 cols)
__global__ void k_site_pre(const float* __restrict__ Sp, const float* __restrict__ Sq,
                           const float* __restrict__ Phi,
                           float* __restrict__ Lp, float* __restrict__ Lq,
                           float* __restrict__ Rm, float* __restrict__ Apad, int sites) {
  int i = blockIdx.x * blockDim.x + threadIdx.x;
  if (i >= sites) return;
  float tmp[9], lp[9], lq[9];

  const float* sp = Sp + 9ull * i;
  for (int k = 0; k < 9; ++k) tmp[k] = sp[k];
  tmp[0] += EPS_REG; tmp[4] += EPS_REG; tmp[8] += EPS_REG;
  inv3(tmp, lp);

  const float* sq = Sq + 9ull * i;
  for (int k = 0; k < 9; ++k) tmp[k] = sq[k];
  tmp[0] += EPS_REG; tmp[4] += EPS_REG; tmp[8] += EPS_REG;
  inv3(tmp, lq);

  // Rodrigues: R = I + s*K + c2*K^2, K = hat(w), K^2 = w w^T - |w|^2 I
  float wx = Phi[3ull*i+0], wy = Phi[3ull*i+1], wz = Phi[3ull*i+2];
  float t2 = wx*wx + wy*wy + wz*wz;
  float s, c2;
  if (t2 < 1e-12f) {
    s  = 1.0f - t2 * (1.0f/6.0f);
    c2 = 0.5f - t2 * (1.0f/24.0f);
  } else {
    float t = sqrtf(t2);
    s  = sinf(t) / t;
    c2 = (1.0f - cosf(t)) / t2;
  }
  float R[9];
  R[0] = 1.0f + c2*(wx*wx - t2); R[1] = -s*wz + c2*wx*wy;       R[2] =  s*wy + c2*wx*wz;
  R[3] =  s*wz + c2*wy*wx;       R[4] = 1.0f + c2*(wy*wy - t2); R[5] = -s*wx + c2*wy*wz;
  R[6] = -s*wy + c2*wz*wx;       R[7] =  s*wx + c2*wz*wy;       R[8] = 1.0f + c2*(wz*wz - t2);

  // A = R^T * Lq * R
  float T[9], A[9];
  mm3(lq, R, T);
  mTm3(R, T, A);

  float* ap = Apad + 16ull * i;
  for (int k = 0; k < 9; ++k) ap[k] = A[k];
  for (int k = 9; k < 16; ++k) ap[k] = 0.0f;
  for (int k = 0; k < 9; ++k) { Lp[9ull*i+k] = lp[k]; Lq[9ull*i+k] = lq[k]; Rm[9ull*i+k] = R[k]; }
}

// ---------- kernel 2: column sums of beta (recoil weights) ----------
__global__ void k_colsum(const float* __restrict__ beta, float* __restrict__ colsum,
                         int Bb, int N) {
  int idx = blockIdx.x * blockDim.x + threadIdx.x;
  if (idx >= Bb * N) return;
  int b = idx / N, col = idx - b * N;
  const float* p = beta + (size_t)b * N * N + col;
  float acc = 0.0f;
  for (int r = 0; r < N; ++r) acc += p[(size_t)r * N];
  colsum[idx] = acc;
}

// ---------- kernel 3: S[b] = beta[b] (NxN) * Apad[b] (Nx16) via f32 WMMA ----------
// One wave per 16-row output tile. EXEC is all-ones inside each active wave.
__global__ void k_gemm_wmma(const float* __restrict__ beta, const float* __restrict__ Apad,
                            float* __restrict__ S, int Bb, int N) {
  const int tilesPerBatch = N >> 4;
  const int numTiles = Bb * tilesPerBatch;
  const int gw   = (blockIdx.x * blockDim.x + threadIdx.x) >> 5;  // uniform per wave
  const int lane = threadIdx.x & 31;
  if (gw >= numTiles) return;

  const int b    = gw / tilesPerBatch;
  const int row0 = (gw - b * tilesPerBatch) << 4;
  const float* __restrict__ betaB = beta + (size_t)b * N * N;
  const float* __restrict__ Ab    = Apad + (size_t)b * N * 16;

  const int half = lane >> 4;      // 0: K,K+1 ; 1: K+2,K+3  (ISA 32-bit A layout)
  const int l    = lane & 15;      // M (A-frag) / N (B,C frags)

  v8f acc = {0.f, 0.f, 0.f, 0.f, 0.f, 0.f, 0.f, 0.f};
  const float* arow = betaB + (size_t)(row0 + l) * N + 2 * half;

  for (int k = 0; k < N; k += 4) {
    // A-frag: lane l holds beta[row0+l][k + 2*half + {0,1}]
    v2f a = *(const v2f*)(arow + k);
    // B-frag: VGPR0 = row (k+2*half), VGPR1 = row (k+2*half+1), N = l
    const int kb = k + 2 * half;
    v2f bf;
    bf.x = Ab[(size_t)kb * 16 + l];
    bf.y = Ab[(size_t)(kb + 1) * 16 + l];
    acc = __builtin_amdgcn_wmma_f32_16x16x4_f32(false, a, false, bf,
                                                (short)0, acc, false, false);
  }

  // C/D layout: VGPR r -> row (row0 + r + 8*half), col l
  for (int r = 0; r < 8; ++r)
    S[((size_t)b * N + row0 + r + 8 * half) * 16 + l] = acc[r];
}

// ---------- eigen helpers (3x3 symmetric) ----------
__device__ __forceinline__ void cross3(const float* a, const float* b, float* c) {
  c[0] = a[1]*b[2] - a[2]*b[1];
  c[1] = a[2]*b[0] - a[0]*b[2];
  c[2] = a[0]*b[1] - a[1]*b[0];
}
__device__ __forceinline__ float dot3(const float* a, const float* b) {
  return a[0]*b[0] + a[1]*b[1] + a[2]*b[2];
}
__device__ void eigvec3(float a00, float a01, float a02, float a11, float a12, float a22,
                        float lam, float* v) {
  float r0[3] = {a00 - lam, a01, a02};
  float r1[3] = {a01, a11 - lam, a12};
  float r2[3] = {a02, a12, a22 - lam};
  float c01[3], c02[3], c12[3];
  cross3(r0, r1, c01); cross3(r0, r2, c02); cross3(r1, r2, c12);
  float n01 = dot3(c01, c01), n02 = dot3(c02, c02), n12 = dot3(c12, c12);
  float* best = c01; float nb = n01;
  if (n02 > nb) { best = c02; nb = n02; }
  if (n12 > nb) { best = c12; nb = n12; }
  if (nb < 1e-30f) { v[0] = 1.0f; v[1] = 0.0f; v[2] = 0.0f; return; }
  float inv = rsqrtf(nb);
  v[0] = best[0]*inv; v[1] = best[1]*inv; v[2] = best[2]*inv;
}

// ---------- kernel 4: assemble M, eigen-floor, outputs ----------
__global__ void k_finalize(const float* __restrict__ Lp, const float* __restrict__ Lq,
                           const float* __restrict__ Rm, const float* __restrict__ S,
                           const float* __restrict__ colsum,
                           float* __restrict__ Mpd, float* __restrict__ Minv, int sites) {
  int i = blockIdx.x * blockDim.x + threadIdx.x;
  if (i >= sites) return;

  float R[9], lq[9], lp[9], Ss[9], T[9], Min[9], M[9];
  for (int k = 0; k < 9; ++k) { R[k] = Rm[9ull*i+k]; lq[k] = Lq[9ull*i+k]; lp[k] = Lp[9ull*i+k]; }
  for (int k = 0; k < 9; ++k) Ss[k] = S[16ull*i + k];
  float cs = colsum[i];

  // M_in = R * Ssum * R^T
  mm3(R, Ss, T);
  mmT3(T, R, Min);
  for (int k = 0; k < 9; ++k) M[k] = lp[k] + Min[k] + cs * lq[k];

  // symmetrize
  float a00 = M[0], a11 = M[4], a22 = M[8];
  float a01 = 0.5f * (M[1] + M[3]);
  float a02 = 0.5f * (M[2] + M[6]);
  float a12 = 0.5f * (M[5] + M[7]);
  float Msym[9] = {a00, a01, a02, a01, a11, a12, a02, a12, a22};

  // closed-form eigenvalues (trig method)
  float p1 = a01*a01 + a02*a02 + a12*a12;
  float q  = (a00 + a11 + a22) * (1.0f/3.0f);
  float d0 = a00 - q, d1 = a11 - q, d2 = a22 - q;
  float p2 = d0*d0 + d1*d1 + d2*d2 + 2.0f * p1;
  float w0, w1, w2;
  if (p2 < 1e-30f) {
    w0 = w1 = w2 = q;
  } else {
    float p = sqrtf(p2 * (1.0f/6.0f));
    float ip = 1.0f / p;
    float b00 = d0*ip, b11 = d1*ip, b22 = d2*ip;
    float b01 = a01*ip, b02 = a02*ip, b12 = a12*ip;
    float detB = b00*(b11*b22 - b12*b12) - b01*(b01*b22 - b12*b02) + b02*(b01*b12 - b11*b02);
    float r = 0.5f * detB;
    r = fminf(1.0f, fmaxf(-1.0f, r));
    float phi_a = acosf(r) * (1.0f/3.0f);
    w2 = q + 2.0f * p * cosf(phi_a);                        // largest
    w0 = q + 2.0f * p * cosf(phi_a + 2.0943951023931953f);  // smallest
    w1 = 3.0f * q - w0 - w2;
  }

  float* mp = Mpd  + 9ull * i;
  float* mi = Minv + 9ull * i;

  if (w0 >= MIN_EIG) {
    // No floor triggered: M_pd = M, M_inv = M^-1 (matches eigh reconstruction in f32)
    float inv[9];
    inv3(Msym, inv);
    for (int k = 0; k < 9; ++k) { mp[k] = Msym[k]; mi[k] = inv[k]; }
  } else {
    // Rare path: full eigen reconstruction with floored spectrum
    float v0[3], v2[3], v1[3];
    eigvec3(a00, a01, a02, a11, a12, a22, w0, v0);
    eigvec3(a00, a01, a02, a11, a12, a22, w2, v2);
    float d = dot3(v2, v0);
    v2[0] -= d*v0[0]; v2[1] -= d*v0[1]; v2[2] -= d*v0[2];
    float nn = dot3(v2, v2);
    if (nn < 1e-12f) {
      float e[3] = {1.0f, 0.0f, 0.0f};
      if (fabsf(v0[0]) > 0.9f) { e[0] = 0.0f; e[1] = 1.0f; }
      cross3(v0, e, v2);
      nn = dot3(v2, v2);
    }
    float inv = rsqrtf(nn);
    v2[0]*=inv; v2[1]*=inv; v2[2]*=inv;
    cross3(v0, v2, v1);

    float wf0 = fmaxf(w0, MIN_EIG), wf1 = fmaxf(w1, MIN_EIG), wf2 = fmaxf(w2, MIN_EIG);
    float g0 = 1.0f/wf0, g1 = 1.0f/wf1, g2 = 1.0f/wf2;
    for (int r = 0; r < 3; ++r)
      for (int c = 0; c < 3; ++c) {
        mp[3*r+c] = wf0*v0[r]*v0[c] + wf1*v1[r]*v1[c] + wf2*v2[r]*v2[c];
        mi[3*r+c] = g0 *v0[r]*v0[c] + g1 *v1[r]*v1[c] + g2 *v2[r]*v2[c];
      }
  }
}

extern "C" void kernel_launch(void* const* d_in, const int* in_sizes, int n_in,
                              void* d_out, int out_size, void* d_ws, size_t ws_size,
                              hipStream_t stream) {
  const float* Sp   = (const float*)d_in[0];
  const float* Sq   = (const float*)d_in[1];
  const float* Phi  = (const float*)d_in[2];
  const float* beta = (const float*)d_in[3];
  // d_in[4] = generators (hat-map basis), folded analytically into Rodrigues.

  const int sites = in_sizes[0] / 9;          // B*N
  const int N     = in_sizes[3] / sites;      // (B*N*N)/(B*N)
  const int Bb    = sites / N;

  float* ws     = (float*)d_ws;
  float* Apad   = ws;                          // sites*16
  float* S      = Apad + (size_t)sites * 16;   // sites*16
  float* Lp     = S    + (size_t)sites * 16;   // sites*9
  float* Lq     = Lp   + (size_t)sites * 9;    // sites*9
  float* Rm     = Lq   + (size_t)sites * 9;    // sites*9
  float* colsum = Rm   + (size_t)sites * 9;    // sites

  float* Mpd  = (float*)d_out;
  float* Minv = Mpd + (size_t)sites * 9;

  const int TB = 256;
  k_site_pre<<<(sites + TB - 1) / TB, TB, 0, stream>>>(Sp, Sq, Phi, Lp, Lq, Rm, Apad, sites);
  k_colsum  <<<(sites + TB - 1) / TB, TB, 0, stream>>>(beta, colsum, Bb, N);

  const int numTiles     = Bb * (N / 16);
  const int totalThreads = numTiles * 32;
  k_gemm_wmma<<<(totalThreads + TB - 1) / TB, TB, 0, stream>>>(beta, Apad, S, Bb, N);

  k_finalize<<<(sites + TB - 1) / TB, TB, 0, stream>>>(Lp, Lq, Rm, S, colsum, Mpd, Minv, sites);
}